// SpGraphAttentionLayer_2138893713529
// MI455X (gfx1250) — compile-verified
//
#include <hip/hip_runtime.h>
#include <hip/hip_bf16.h>
#include <math.h>

typedef __attribute__((ext_vector_type(2))) float v2f;
typedef __attribute__((ext_vector_type(8))) float v8f;

#define B_ 4
#define N_ 50000
#define E_ 800000
#define FIN_ 128
#define FOUT_ 64
#define ALPHA_ 0.2f
#define NEG_BIG_ (-9000000000000000.0f)

#define ROWS_ (B_ * N_)          // 200000 rows of h
#define TILES_ (ROWS_ / 16)      // 12500 row tiles (exact)

// -------------------------------------------------------------------------
// Kernel 1: h = X @ W using V_WMMA_F32_16X16X4_F32 (fp32 in/out, matches ref)
// One wave computes a 16-row x 64-col tile: 4 v8f accumulators, K=128 in
// steps of 4 -> 32 k-iters x 4 n-tiles = 128 WMMAs per wave.
// -------------------------------------------------------------------------
__global__ __launch_bounds__(256) void gat_gemm_wmma(const float* __restrict__ X,
                                                     const float* __restrict__ W,
                                                     float* __restrict__ h) {
    const int lane   = threadIdx.x & 31;
    const int waveId = threadIdx.x >> 5;
    const int rt     = blockIdx.x * 8 + waveId;   // row-tile index (wave-uniform)
    if (rt >= TILES_) return;                      // uniform per wave: EXEC stays all-1s

    const int m_base = rt * 16;
    const int mrow   = m_base + (lane & 15);       // A-row this lane loads
    const int khalf  = (lane >> 4) << 1;           // lanes 0-15 -> K+{0,1}, 16-31 -> K+{2,3}
    const int ncol   = lane & 15;

    v8f acc0 = {}, acc1 = {}, acc2 = {}, acc3 = {};

    for (int k = 0; k < FIN_; k += 4) {
        const int kk = k + khalf;
        v2f a;
        a.x = X[(size_t)mrow * FIN_ + kk];
        a.y = X[(size_t)mrow * FIN_ + kk + 1];

        const float* wr0 = W + (size_t)kk * FOUT_;
        const float* wr1 = wr0 + FOUT_;

        v2f b0, b1, b2, b3;
        b0.x = wr0[ncol];      b0.y = wr1[ncol];
        b1.x = wr0[16 + ncol]; b1.y = wr1[16 + ncol];
        b2.x = wr0[32 + ncol]; b2.y = wr1[32 + ncol];
        b3.x = wr0[48 + ncol]; b3.y = wr1[48 + ncol];

        acc0 = __builtin_amdgcn_wmma_f32_16x16x4_f32(false, a, false, b0, (short)0, acc0, false, false);
        acc1 = __builtin_amdgcn_wmma_f32_16x16x4_f32(false, a, false, b1, (short)0, acc1, false, false);
        acc2 = __builtin_amdgcn_wmma_f32_16x16x4_f32(false, a, false, b2, (short)0, acc2, false, false);
        acc3 = __builtin_amdgcn_wmma_f32_16x16x4_f32(false, a, false, b3, (short)0, acc3, false, false);
    }

    // C/D layout: VGPR i -> row = i + 8*(lane/16), col = lane%16 (+16*tile)
    const int rhalf = (lane >> 4) << 3;
#pragma unroll
    for (int i = 0; i < 8; ++i) {
        const size_t row = (size_t)(m_base + i + rhalf);
        h[row * FOUT_ +  0 + ncol] = acc0[i];
        h[row * FOUT_ + 16 + ncol] = acc1[i];
        h[row * FOUT_ + 32 + ncol] = acc2[i];
        h[row * FOUT_ + 48 + ncol] = acc3[i];
    }
}

// -------------------------------------------------------------------------
// Kernel 2: per-row attention scores e1 = h@a1, e2 = h@a2
// -------------------------------------------------------------------------
__global__ __launch_bounds__(256) void gat_scores(const float* __restrict__ h,
                                                  const float* __restrict__ a,
                                                  float* __restrict__ e1,
                                                  float* __restrict__ e2) {
    const int row = blockIdx.x * blockDim.x + threadIdx.x;
    if (row >= ROWS_) return;
    const float* hr = h + (size_t)row * FOUT_;
    float s1 = 0.f, s2 = 0.f;
#pragma unroll
    for (int f = 0; f < FOUT_; ++f) {
        const float hv = hr[f];
        s1 += hv * a[f];
        s2 += hv * a[FOUT_ + f];
    }
    e1[row] = s1;
    e2[row] = s2;
}

// -------------------------------------------------------------------------
// Kernel 3: edge scatter. 64 threads per edge (f = feature), 4 edges per
// 256-thread block. ee broadcast through LDS; coalesced h[dst] reads and
// coalesced f32 atomics into the accumulator (L2-resident, 51 MB).
// -------------------------------------------------------------------------
__global__ __launch_bounds__(256) void gat_edges(const int* __restrict__ edges,
                                                 const float* __restrict__ h,
                                                 const float* __restrict__ e1,
                                                 const float* __restrict__ e2,
                                                 float* __restrict__ accum,
                                                 float* __restrict__ rowsum) {
    __shared__ float s_ee[4];
    __shared__ int   s_src[4];
    __shared__ int   s_dst[4];

    const int tid = threadIdx.x;
    const int le  = tid >> 6;   // local edge 0..3
    const int f   = tid & 63;   // feature

    const long long eg = (long long)blockIdx.x * 4 + le;
    const bool valid = eg < (long long)B_ * E_;
    int b = 0, e = 0;
    if (valid) {
        b = (int)(eg / E_);
        e = (int)(eg % E_);
        if (f == 0) {
            const int src = edges[((size_t)b * 2 + 0) * E_ + e];
            const int dst = edges[((size_t)b * 2 + 1) * E_ + e];
            const float s  = e1[(size_t)b * N_ + src] + e2[(size_t)b * N_ + dst];
            const float lr = (s >= 0.f) ? s : ALPHA_ * s;
            const float ee = __expf(-lr);
            s_ee[le]  = ee;
            s_src[le] = src;
            s_dst[le] = dst;
            atomicAdd(&rowsum[(size_t)b * N_ + src], ee);
        }
    }
    __syncthreads();
    if (valid) {
        const int   src = s_src[le];
        const int   dst = s_dst[le];
        const float ee  = s_ee[le];
        const float hv  = h[((size_t)b * N_ + dst) * FOUT_ + f];
        atomicAdd(&accum[((size_t)b * N_ + src) * FOUT_ + f], ee * hv);
    }
}

// -------------------------------------------------------------------------
// Kernel 4: finalize: divide, NaN -> NEG_BIG, ELU
// -------------------------------------------------------------------------
__global__ __launch_bounds__(256) void gat_finalize(const float* __restrict__ accum,
                                                    const float* __restrict__ rowsum,
                                                    float* __restrict__ out) {
    const long long idx = (long long)blockIdx.x * blockDim.x + threadIdx.x;
    if (idx >= (long long)ROWS_ * FOUT_) return;
    const long long row = idx >> 6;
    float v = accum[idx] / rowsum[row];
    if (isnan(v)) v = NEG_BIG_;
    out[idx] = (v > 0.f) ? v : expm1f(v);
}

extern "C" void kernel_launch(void* const* d_in, const int* in_sizes, int n_in,
                              void* d_out, int out_size, void* d_ws, size_t ws_size,
                              hipStream_t stream) {
    const float* X     = (const float*)d_in[0];   // (B, N, FIN)
    const int*   edges = (const int*)d_in[1];     // (B, 2, E)
    const float* W     = (const float*)d_in[2];   // (FIN, FOUT)
    const float* a     = (const float*)d_in[3];   // (1, 2*FOUT)
    float*       out   = (float*)d_out;           // (B, N, FOUT)

    // Workspace layout (floats): accum | rowsum | h | e1 | e2  (~105 MB)
    float* accum  = (float*)d_ws;                       // B*N*FOUT
    float* rowsum = accum + (size_t)ROWS_ * FOUT_;      // B*N
    float* h      = rowsum + (size_t)ROWS_;             // B*N*FOUT
    float* e1     = h + (size_t)ROWS_ * FOUT_;          // B*N
    float* e2     = e1 + (size_t)ROWS_;                 // B*N

    // Zero accumulators (accum + rowsum contiguous)
    hipMemsetAsync(accum, 0, ((size_t)ROWS_ * FOUT_ + ROWS_) * sizeof(float), stream);

    // 1) h = X @ W  (WMMA fp32)
    {
        const int waves_per_block = 8;
        const int blocks = (TILES_ + waves_per_block - 1) / waves_per_block;
        gat_gemm_wmma<<<blocks, 256, 0, stream>>>(X, W, h);
    }
    // 2) scores
    {
        const int blocks = (ROWS_ + 255) / 256;
        gat_scores<<<blocks, 256, 0, stream>>>(h, a, e1, e2);
    }
    // 3) edge scatter: 4 edges per block
    {
        const long long total_edges = (long long)B_ * E_;
        const int blocks = (int)((total_edges + 3) / 4);
        gat_edges<<<blocks, 256, 0, stream>>>(edges, h, e1, e2, accum, rowsum);
    }
    // 4) finalize
    {
        const long long total = (long long)ROWS_ * FOUT_;
        const int blocks = (int)((total + 255) / 256);
        gat_finalize<<<blocks, 256, 0, stream>>>(accum, rowsum, out);
    }
}